// Fiery_78486232367648
// MI455X (gfx1250) — compile-verified
//
#include <hip/hip_runtime.h>
#include <hip/hip_bf16.h>
#include <stdint.h>

// ---------------------------------------------------------------------------
// FIERY BEV voxel pooling for MI455X (gfx1250).
// x:        [B=2, N=6, D=48, H=28, W=60, C=64] f32  -> flat [P=967680, 64]
// geometry: [B, N, D, H, W, 3] f32                  -> flat [P, 3]
// out:      [B, 64, 200, 200] f32
//
// Memory-bound scatter-add (0.22 flop/byte -> WMMA inapplicable). CDNA5 paths:
//   * global_load_async_to_lds_b128 double-buffered pipeline (ASYNCcnt)
//   * s_wait_asynccnt partial waits (in-order async load completion)
//   * single-instruction no-return global_atomic_add_f32 (forced via asm,
//     base VGPR pair + offset: immediates for the 16-channel burst)
// ---------------------------------------------------------------------------

#define NPTS        967680      // B*N*D*H*W
#define PTS_PER_B   483840
#define NVOX        40000       // 200*200
#define CCH         64
#define TP          64          // points per async tile (64*256B = 16KB)
#define NTILES      (NPTS / TP)               // 15120
#define SCAT_BLOCKS 504
#define TILES_PER_BLK (NTILES / SCAT_BLOCKS)  // 30

// no-return hardware fp32 atomic: global_atomic_add_f32 v[a:b], vdata, off
#define ATOM(OFS, VAL)                                                \
    asm volatile("global_atomic_add_f32 %0, %1, off offset:" #OFS    \
                 :: "v"(dst), "v"(VAL) : "memory")

__global__ __launch_bounds__(256)
void Fiery_scatter(const float* __restrict__ x,
                   const float* __restrict__ geo,
                   float* __restrict__ ws) {
    __shared__ float4 buf[2][TP * 16];          // 2 x 16KB double buffer
    const int tid = threadIdx.x;
    const int t0  = blockIdx.x * TILES_PER_BLK;

    // Issue one tile's 16KB as 4 async b128 transfers per lane (per-wave
    // ASYNCcnt += 4). LDS dest offset mirrors global offset, so LDS float4
    // index == in-tile float4 index.
    auto issue = [&](int tile, int bsel) {
        const char* gsrc = (const char*)x + (size_t)tile * (TP * CCH * 4);
        uint32_t lds0 = (uint32_t)(size_t)(&buf[bsel][0]);
#pragma unroll
        for (int i = 0; i < 4; ++i) {
            uint32_t off = (uint32_t)(tid * 16 + i * 4096);
            const char* ga = gsrc + off;
            uint32_t la = lds0 + off;
            asm volatile("global_load_async_to_lds_b128 %0, %1, off"
                         :: "v"(la), "v"(ga) : "memory");
        }
    };

    issue(t0, 0);                                // prefetch first tile

    for (int i = 0; i < TILES_PER_BLK; ++i) {
        const int tile = t0 + i;
        const int cur  = i & 1;

        if (i + 1 < TILES_PER_BLK) {
            issue(tile + 1, cur ^ 1);            // overlap next tile's DMA
            // async loads complete in order: <=4 outstanding => current done
            asm volatile("s_wait_asynccnt 0x4" ::: "memory");
        } else {
            asm volatile("s_wait_asynccnt 0x0" ::: "memory");
        }
        __syncthreads();                         // tile visible to all waves

        // 4 lanes per point; lane handles 16 channels (4 x float4)
        const int pidx = tid >> 2;               // 0..63 point in tile
        const int sub  = tid & 3;                // channel group
        const int p    = tile * TP + pidx;       // global point id
        const int bb   = (p >= PTS_PER_B) ? 1 : 0;

        const float* g = geo + (size_t)p * 3;
        // exact reference math: ((geo - (start - res/2)) / res) as int32
        // start-res/2 = (-50,-50,-10), res = (0.5,0.5,20); trunc toward zero
        float fx = (g[0] + 50.0f) / 0.5f;
        float fy = (g[1] + 50.0f) / 0.5f;
        float fz = (g[2] + 10.0f) / 20.0f;
        int vx = (int)fx, vy = (int)fy, vz = (int)fz;
        bool valid = (vx >= 0) & (vx < 200) & (vy >= 0) & (vy < 200) & (vz == 0);

        if (valid) {
            const uint32_t rank = (uint32_t)(vx * 200 + vy);
            float* dst = ws + ((size_t)bb * NVOX + rank) * CCH + sub * 16;
            const float4* s4 = &buf[cur][pidx * 16 + sub * 4];
            float4 v0 = s4[0], v1 = s4[1], v2 = s4[2], v3 = s4[3];
            ATOM(0,  v0.x); ATOM(4,  v0.y); ATOM(8,  v0.z); ATOM(12, v0.w);
            ATOM(16, v1.x); ATOM(20, v1.y); ATOM(24, v1.z); ATOM(28, v1.w);
            ATOM(32, v2.x); ATOM(36, v2.y); ATOM(40, v2.z); ATOM(44, v2.w);
            ATOM(48, v3.x); ATOM(52, v3.y); ATOM(56, v3.z); ATOM(60, v3.w);
        }
        __syncthreads();                         // buffer reuse fence
    }
}

__global__ void Fiery_zero_ws(float4* __restrict__ ws) {
    // 2*40000*64 floats = 1,280,000 float4 ; grid = 5000 x 256
    size_t i = (size_t)blockIdx.x * blockDim.x + threadIdx.x;
    ws[i] = make_float4(0.f, 0.f, 0.f, 0.f);
}

__global__ __launch_bounds__(256)
void Fiery_transpose(const float* __restrict__ ws, float* __restrict__ out) {
    // ws[b][r][c] -> out[b][c][r], r in [0,40000), 64-row tiles
    __shared__ float t[64][65];                  // stride-65: conflict-free
    const int blk = blockIdx.x;                  // 2 * 625
    const int b   = blk / 625;
    const int r0  = (blk % 625) * 64;
    const int tid = threadIdx.x;

    const float* src = ws + ((size_t)b * NVOX + r0) * CCH;
#pragma unroll
    for (int i = 0; i < 16; ++i) {               // coalesced along c
        int idx = i * 256 + tid;                 // 0..4095
        t[idx >> 6][idx & 63] = src[idx];
    }
    __syncthreads();
    float* dst = out + (size_t)b * CCH * NVOX + r0;
#pragma unroll
    for (int i = 0; i < 16; ++i) {               // coalesced along r
        int idx = i * 256 + tid;
        int c = idx >> 6, r = idx & 63;
        dst[(size_t)c * NVOX + r] = t[r][c];
    }
}

extern "C" void kernel_launch(void* const* d_in, const int* in_sizes, int n_in,
                              void* d_out, int out_size, void* d_ws, size_t ws_size,
                              hipStream_t stream) {
    (void)in_sizes; (void)n_in; (void)out_size; (void)ws_size;
    const float* x   = (const float*)d_in[0];
    const float* geo = (const float*)d_in[1];
    float* out = (float*)d_out;
    float* ws  = (float*)d_ws;   // needs 2*40000*64*4 = 20.48 MB

    Fiery_zero_ws<<<5000, 256, 0, stream>>>((float4*)ws);
    Fiery_scatter<<<SCAT_BLOCKS, 256, 0, stream>>>(x, geo, ws);
    Fiery_transpose<<<1250, 256, 0, stream>>>(ws, out);
}